// PDFSampler_7928509628624
// MI455X (gfx1250) — compile-verified
//
#include <hip/hip_runtime.h>

// PDF importance sampler (NeRF sample_pdf style), fused single kernel.
// One wave32 per ray; cumsum via V_WMMA_F32_16X16X4_F32 (P = L16 * W).

#define HIST_PAD 0.01f
#define EPS_PAD  1e-5f
#define NW   128   // coarse samples per ray
#define NB   129   // existing bins == num output samples (num_samples+1)
#define NOUT 258   // merged output bins per ray

typedef __attribute__((ext_vector_type(2))) float v2f;
typedef __attribute__((ext_vector_type(8))) float v8f;

// per-wave LDS slice layout (floats): cdf[132] | bins[132] | smp[132] | stg[260]
#define SLICE 656

__global__ __launch_bounds__(256)
void pdf_sampler_kernel(const float* __restrict__ weights,
                        const float* __restrict__ existing_bins,
                        const float* __restrict__ near_p,
                        const float* __restrict__ far_p,
                        float* __restrict__ out,
                        int R)
{
    const int lane = threadIdx.x & 31;
    const int wave = threadIdx.x >> 5;
    const int ray  = blockIdx.x * 8 + wave;

    __shared__ float smem[8 * SLICE];
    float* cdf  = smem + wave * SLICE;   // 129 used
    float* bins = cdf  + 132;            // 129 used
    float* smp  = bins + 132;            // 129 used
    float* stg  = smp  + 132;            // 258 used

    if (ray >= R) return;                // wave-uniform: EXEC stays all-ones

    const float nearv = *near_p;
    const float farv  = *far_p;
    const float span  = farv - nearv;

    // ---- load 128 weights, coalesced float4 (ray*512B is 16B aligned) ----
    const float4* wp = (const float4*)(weights + (size_t)ray * NW);
    float4 wv = wp[lane];
    wv.x += HIST_PAD; wv.y += HIST_PAD; wv.z += HIST_PAD; wv.w += HIST_PAD;

    // ---- wave reduction for w_sum, then reference's EPS padding path ----
    float lsum = wv.x + wv.y + wv.z + wv.w;
    #pragma unroll
    for (int d = 16; d > 0; d >>= 1) lsum += __shfl_xor(lsum, d, 32);
    const float wsum0   = lsum;
    const float padding = fmaxf(EPS_PAD - wsum0, 0.0f);
    const float padEach = padding * (1.0f / NW);
    wv.x += padEach; wv.y += padEach; wv.z += padEach; wv.w += padEach;
    const float inv_wsum = 1.0f / (wsum0 + padding);

    // ---- stage existing_bins into LDS (coalesced scalar loads) ----
    const float* ebase = existing_bins + (size_t)ray * NB;
    bins[lane]      = ebase[lane];
    bins[lane + 32] = ebase[lane + 32];
    bins[lane + 64] = ebase[lane + 64];
    bins[lane + 96] = ebase[lane + 96];
    if (lane == 0) bins[128] = ebase[128];

    // ---- blocked prefix sum via WMMA: P(16x16) = L16(16x16) x W(16x16) ----
    // W[r][j] = w[16*j + r] for chunk j<8, zero for j>=8. 4 chained K=4 WMMAs.
    const int half = lane >> 4;          // K-half for A/B; row-half for C/D
    const int m    = lane & 15;          // A: M row. B: N col. C/D: N col.
    v8f acc = {};
    #pragma unroll
    for (int k = 0; k < 4; ++k) {
        v2f a, b;
        const int k0 = 4 * k + 2 * half; // global K index of a.x
        a.x = (k0     <= m) ? 1.0f : 0.0f;   // L16[m][k0]
        a.y = (k0 + 1 <= m) ? 1.0f : 0.0f;   // L16[m][k0+1]
        // B[kk][j]: lanes<16 carry K=4k+{0,1} (wv.x,wv.y of lane 4j+k),
        //           lanes>=16 carry K=4k+{2,3} (wv.z,wv.w of lane 4j+k)
        const int src = 4 * m + k;
        const float bxl = __shfl(wv.x, src, 32);
        const float byl = __shfl(wv.y, src, 32);
        const float bxh = __shfl(wv.z, src, 32);
        const float byh = __shfl(wv.w, src, 32);
        b.x = (m < 8) ? (half ? bxh : bxl) : 0.0f;
        b.y = (m < 8) ? (half ? byh : byl) : 0.0f;
        acc = __builtin_amdgcn_wmma_f32_16x16x4_f32(
            false, a, false, b, (short)0, acc, false, false);
    }

    // chunk totals live in lanes 16..23, acc[7]; exclusive scan of 8 values
    const float ctot = acc[7];
    float off = 0.0f;
    #pragma unroll
    for (int i = 0; i < 8; ++i) {
        const float ti = __shfl(ctot, 16 + i, 32);
        off += (i < m) ? ti : 0.0f;
    }

    // cdf[16*m + 8*half + r + 1] = min(1, (P + off) / w_sum); cdf[0] = 0
    if (m < 8) {
        const int base = 16 * m + 8 * half;
        #pragma unroll
        for (int r = 0; r < 8; ++r) {
            const float v = (acc[r] + off) * inv_wsum;
            cdf[base + r + 1] = fminf(v, 1.0f);
        }
    }
    if (lane == 0) cdf[0] = 0.0f;
    __syncthreads();

    // ---- invert CDF at 129 midpoint quantiles (searchsorted right + lerp) ----
    for (int j = lane; j < NB; j += 32) {
        const float u = ((float)j + 0.5f) * (1.0f / 129.0f);
        int lo = 0, hi = NB;
        while (lo < hi) {                 // upper_bound over cdf[0..128]
            const int mid = (lo + hi) >> 1;
            if (cdf[mid] <= u) lo = mid + 1; else hi = mid;
        }
        const int below = lo - 1;                     // lo>=1 since cdf[0]=0<u
        const int above = (lo > NW) ? NW : lo;
        const float c0 = cdf[below], c1 = cdf[above];
        const float b0 = bins[below], b1 = bins[above];
        const float denom = c1 - c0;
        float t;
        if (denom != 0.0f) {
            t = fminf(fmaxf((u - c0) / denom, 0.0f), 1.0f);
        } else {
            t = (u > c0) ? 1.0f : 0.0f;   // matches nan_to_num(+inf/nan)+clip
        }
        smp[j] = b0 + t * (b1 - b0);      // sorted, since u and cdf monotone
    }
    __syncthreads();

    // ---- merge two sorted 129-lists by rank (replaces jnp.sort of 258) ----
    for (int i = lane; i < NB; i += 32) {
        const float v = bins[i];
        int lo = 0, hi = NB;              // rank = #samples strictly < v
        while (lo < hi) { const int mid = (lo + hi) >> 1;
                          if (smp[mid] < v) lo = mid + 1; else hi = mid; }
        stg[i + lo] = fmaf(span, v, nearv);
    }
    for (int j = lane; j < NB; j += 32) {
        const float v = smp[j];
        int lo = 0, hi = NB;              // rank = #bins <= v (tie: bins first)
        while (lo < hi) { const int mid = (lo + hi) >> 1;
                          if (bins[mid] <= v) lo = mid + 1; else hi = mid; }
        stg[j + lo] = fmaf(span, v, nearv);
    }
    __syncthreads();

    // ---- coalesced store of the merged euclidean bins ----
    float* orow = out + (size_t)ray * NOUT;
    for (int k = lane; k < NOUT; k += 32) orow[k] = stg[k];
}

extern "C" void kernel_launch(void* const* d_in, const int* in_sizes, int n_in,
                              void* d_out, int out_size, void* d_ws, size_t ws_size,
                              hipStream_t stream) {
    (void)n_in; (void)out_size; (void)d_ws; (void)ws_size;
    const float* weights = (const float*)d_in[0];
    const float* ebins   = (const float*)d_in[1];
    const float* nearp   = (const float*)d_in[2];
    const float* farp    = (const float*)d_in[3];
    // d_in[4] (num_samples) is structurally fixed at 128 by the shapes.
    float* out = (float*)d_out;

    const int R = in_sizes[0] / NW;          // 65536
    const int blocks = (R + 7) / 8;          // 8 rays (waves) per 256-thr block
    hipLaunchKernelGGL(pdf_sampler_kernel, dim3(blocks), dim3(256), 0, stream,
                       weights, ebins, nearp, farp, out, R);
}